// TwhinModel_18021682774350
// MI455X (gfx1250) — compile-verified
//
#include <hip/hip_runtime.h>

// ---------------------------------------------------------------------------
// TwhinModel logits + sigmoid for MI455X (gfx1250, wave32).
//   B=8192, D=256, R=8, N=1024, C=1024, F=2
//   out[0 .. TOT)      = logits  (pos | negs_lhs/negs_rhs per relation)
//   out[TOT .. 2*TOT)  = sigmoid(logits)
// Workspace: needs 2*B*D*4 = 16 MiB (lhs array then rhs array).
// ---------------------------------------------------------------------------

typedef float v2f __attribute__((ext_vector_type(2)));
typedef float v8f __attribute__((ext_vector_type(8)));

constexpr int kB = 8192;
constexpr int kD = 256;
constexpr int kR = 8;
constexpr int kN = 1024;
constexpr int kC = 1024;                     // kB / kR
constexpr long long kTot = (long long)kB + 2LL * kR * kC * kN;  // 16,785,408

__device__ __forceinline__ float fast_sigmoid(float x) {
    return 1.0f / (1.0f + __expf(-x));
}

// ---------------------------------------------------------------------------
// Kernel 1: x = emb.sum(axis=1); split even rows -> lhs, odd rows -> rhs.
// emb is (2B, 2, D) fp32.  One float4 per thread.
// ---------------------------------------------------------------------------
__global__ __launch_bounds__(256) void fuse_emb_kernel(
    const float* __restrict__ emb, float* __restrict__ lhs, float* __restrict__ rhs) {
    const int p  = blockIdx.x * 256 + threadIdx.x;   // float4 id, 2B*D/4 total
    const int i  = p >> 6;                           // emb row (D/4 == 64)
    const int d  = (p & 63) << 2;                    // float offset in row
    const float4 e0 = *(const float4*)(emb + (size_t)i * (2 * kD) + d);
    const float4 e1 = *(const float4*)(emb + (size_t)i * (2 * kD) + kD + d);
    float4 s;
    s.x = e0.x + e1.x; s.y = e0.y + e1.y; s.z = e0.z + e1.z; s.w = e0.w + e1.w;
    float* dst = (i & 1) ? rhs : lhs;
    *(float4*)(dst + (size_t)(i >> 1) * kD + d) = s;
}

// ---------------------------------------------------------------------------
// Kernel 2: pos[i] = dot(lhs[i], rhs[i] + trans[rels[i]]).  One wave32 / row.
// ---------------------------------------------------------------------------
__global__ __launch_bounds__(256) void pos_kernel(
    const float* __restrict__ lhs, const float* __restrict__ rhs,
    const float* __restrict__ trans, const int* __restrict__ rels,
    float* __restrict__ out) {
    const int gid  = blockIdx.x * 256 + threadIdx.x;
    const int row  = gid >> 5;
    const int lane = gid & 31;
    const float* lp = lhs + (size_t)row * kD + lane * 8;
    const float* rp = rhs + (size_t)row * kD + lane * 8;
    const float* tp = trans + (size_t)rels[row] * kD + lane * 8;
    const float4 l0 = *(const float4*)lp,      l1 = *(const float4*)(lp + 4);
    const float4 r0 = *(const float4*)rp,      r1 = *(const float4*)(rp + 4);
    const float4 t0 = *(const float4*)tp,      t1 = *(const float4*)(tp + 4);
    float acc = l0.x * (r0.x + t0.x) + l0.y * (r0.y + t0.y)
              + l0.z * (r0.z + t0.z) + l0.w * (r0.w + t0.w)
              + l1.x * (r1.x + t1.x) + l1.y * (r1.y + t1.y)
              + l1.z * (r1.z + t1.z) + l1.w * (r1.w + t1.w);
#pragma unroll
    for (int m = 16; m > 0; m >>= 1) acc += __shfl_xor(acc, m, 32);
    if (lane == 0) {
        out[row] = acc;
        out[(size_t)kTot + row] = fast_sigmoid(acc);
    }
}

// ---------------------------------------------------------------------------
// Kernel 3: per-relation gathered GEMMs via V_WMMA_F32_16X16X4_F32.
//   mode 0: negs_lhs = s_rhs (C x D) . samp_lhs^T (D x N)
//   mode 1: negs_rhs = s_lhs (C x D) . samp_rhs^T (D x N)
// Block = 256 threads (8 waves), output tile 128x128; wave -> 32x64 (2x4 WMMA).
// K staged in LDS in chunks of 32 (row stride 36 floats => conflict-free).
// ---------------------------------------------------------------------------
__global__ __launch_bounds__(256) void neg_gemm_kernel(
    const float* __restrict__ lhs, const float* __restrict__ rhs,
    const int* __restrict__ order, const int* __restrict__ negL,
    const int* __restrict__ negR, float* __restrict__ out) {
    constexpr int TILE = 128;
    constexpr int KB   = 32;
    constexpr int LDK  = KB + 4;     // padded stride (36 floats)

    __shared__ float As[TILE][LDK];
    __shared__ float Bs[TILE][LDK];
    __shared__ int   idxA[TILE];
    __shared__ int   idxB[TILE];

    const int cTile = blockIdx.x;
    const int nTile = blockIdx.y;
    const int r     = blockIdx.z >> 1;
    const int mode  = blockIdx.z & 1;
    const int t     = threadIdx.x;

    const float* Aarr = mode ? lhs : rhs;
    const float* Barr = mode ? rhs : lhs;
    const int*   nIdx = mode ? negR : negL;

    if (t < TILE) idxA[t]        = order[r * kC + cTile * TILE + t];
    else          idxB[t - TILE] = nIdx[r * kN + nTile * TILE + (t - TILE)];
    __syncthreads();

    const int w    = t >> 5;
    const int lane = t & 31;
    const int wm   = w & 3;          // 4 wave-rows  * 32 rows = 128
    const int wn   = w >> 2;         // 2 wave-cols  * 64 cols = 128
    const int hl   = lane >> 4;      // lane half (0/1)
    const int mr   = lane & 15;      // row/col within 16x16 tile
    const int koff = hl << 1;        // K sub-offset: 0 or 2

    const v8f vzero = {0.f, 0.f, 0.f, 0.f, 0.f, 0.f, 0.f, 0.f};
    v8f acc[2][4];
#pragma unroll
    for (int i = 0; i < 2; ++i)
#pragma unroll
        for (int j = 0; j < 4; ++j) acc[i][j] = vzero;

    for (int k0 = 0; k0 < kD; k0 += KB) {
        // cooperative gather of 128x32 A and B chunks (4 float4 per thread each)
#pragma unroll
        for (int j = 0; j < 4; ++j) {
            const int q   = t + 256 * j;       // 0..1023
            const int row = q >> 3;
            const int c4  = (q & 7) << 2;
            *(float4*)(&As[row][c4]) =
                *(const float4*)(Aarr + (size_t)idxA[row] * kD + k0 + c4);
            *(float4*)(&Bs[row][c4]) =
                *(const float4*)(Barr + (size_t)idxB[row] * kD + k0 + c4);
            if (k0 + KB < kD) {   // software prefetch next K chunk -> global_prefetch_b8
                __builtin_prefetch(Aarr + (size_t)idxA[row] * kD + k0 + KB + c4, 0, 3);
                __builtin_prefetch(Barr + (size_t)idxB[row] * kD + k0 + KB + c4, 0, 3);
            }
        }
        __syncthreads();

#pragma unroll
        for (int kk = 0; kk < KB; kk += 4) {
            // A frag (16x4): lanes 0-15 -> K{0,1}, lanes 16-31 -> K{2,3}, M = lane&15
            v2f a[2], b[4];
#pragma unroll
            for (int tm = 0; tm < 2; ++tm) {
                const float2 f = *(const float2*)(&As[wm * 32 + tm * 16 + mr][kk + koff]);
                a[tm][0] = f.x; a[tm][1] = f.y;
            }
            // B frag (4x16): VGPR0/1 hold K{0,1} (lanes 0-15) / K{2,3} (lanes 16-31), N = lane&15
#pragma unroll
            for (int tn = 0; tn < 4; ++tn) {
                const float2 f = *(const float2*)(&Bs[wn * 64 + tn * 16 + mr][kk + koff]);
                b[tn][0] = f.x; b[tn][1] = f.y;
            }
#pragma unroll
            for (int tm = 0; tm < 2; ++tm)
#pragma unroll
                for (int tn = 0; tn < 4; ++tn)
                    acc[tm][tn] = __builtin_amdgcn_wmma_f32_16x16x4_f32(
                        false, a[tm], false, b[tn], (short)0, acc[tm][tn],
                        false, false);
        }
        __syncthreads();
    }

    // epilogue: C/D layout: VGPR v -> M = v + 8*half, N = lane&15
    const size_t base = (size_t)kB + (size_t)(r * 2 + mode) * ((size_t)kC * kN);
#pragma unroll
    for (int tm = 0; tm < 2; ++tm)
#pragma unroll
        for (int tn = 0; tn < 4; ++tn)
#pragma unroll
            for (int v = 0; v < 8; ++v) {
                const int cg = cTile * TILE + wm * 32 + tm * 16 + v + hl * 8;
                const int ng = nTile * TILE + wn * 64 + tn * 16 + mr;
                const float x = acc[tm][tn][v];
                const size_t o = base + (size_t)cg * kN + ng;
                out[o] = x;
                out[(size_t)kTot + o] = fast_sigmoid(x);
            }
}

// ---------------------------------------------------------------------------
extern "C" void kernel_launch(void* const* d_in, const int* in_sizes, int n_in,
                              void* d_out, int out_size, void* d_ws, size_t ws_size,
                              hipStream_t stream) {
    const float* emb   = (const float*)d_in[0];
    const float* trans = (const float*)d_in[1];
    const int*   rels  = (const int*)d_in[2];
    const int*   order = (const int*)d_in[3];
    const int*   negL  = (const int*)d_in[4];
    const int*   negR  = (const int*)d_in[5];
    float* out = (float*)d_out;

    float* lhs = (float*)d_ws;                 // B*D floats
    float* rhs = lhs + (size_t)kB * kD;        // B*D floats (16 MiB total)

    // 1) F-axis reduce + lhs/rhs split: 2B*D/4 float4s, 256 threads each
    fuse_emb_kernel<<<(2 * kB * kD / 4) / 256, 256, 0, stream>>>(emb, lhs, rhs);

    // 2) positive logits: one wave32 per row
    pos_kernel<<<(kB * 32) / 256, 256, 0, stream>>>(lhs, rhs, trans, rels, out);

    // 3) 16 gathered GEMMs: grid (C/128, N/128, 2*R)
    dim3 grid(kC / 128, kN / 128, 2 * kR);
    neg_gemm_kernel<<<grid, 256, 0, stream>>>(lhs, rhs, order, negL, negR, out);
}